// SpatioTemporalLinearizedAttention_28621662060945
// MI455X (gfx1250) — compile-verified
//
#include <hip/hip_runtime.h>
#include <hip/hip_bf16.h>

typedef float v2f __attribute__((ext_vector_type(2)));
typedef float v8f __attribute__((ext_vector_type(8)));

constexpr int BB = 16, TT = 12, NN = 512, DD = 128;
constexpr int RR = BB * TT * NN;   // 98304 rows
constexpr int BT = BB * TT;        // 192 (b,t) slabs
constexpr float EPS = 1e-12f;

// ---------------- WMMA helpers (fp32, 16x16x4) ----------------

__device__ __forceinline__ v8f wmma_f32(v2f a, v2f b, v8f c) {
  // (neg_a, A, neg_b, B, c_mod, C, reuse_a, reuse_b)
  return __builtin_amdgcn_wmma_f32_16x16x4_f32(false, a, false, b, (short)0, c,
                                               false, false);
}

// A-fragment: logical A[16 x 4] read from row-major S (row stride ld), origin (row0, k0).
// ISA layout: lanes 0-15 -> M=lane, K={k0,k0+1}; lanes 16-31 -> M=lane-16, K={k0+2,k0+3}.
__device__ __forceinline__ v2f frag_rm(const float* S, int ld, int row0, int k0, int lane) {
  int m  = lane & 15;
  int kb = k0 + ((lane >> 4) << 1);
  const float* p = S + (row0 + m) * ld + kb;
  v2f f; f.x = p[0]; f.y = p[1];
  return f;
}

// B-fragment: logical B[4 x 16] read from row-major S[k][n] (row stride ld), origin (k0, n0).
// Also serves as transposed-A loader: A[m][k] = S[k][m0+m]. Lane-consecutive -> coalesced.
__device__ __forceinline__ v2f frag_cm(const float* S, int ld, int k0, int n0, int lane) {
  int n  = lane & 15;
  int kb = k0 + ((lane >> 4) << 1);
  v2f f;
  f.x = S[kb * ld + n0 + n];
  f.y = S[(kb + 1) * ld + n0 + n];
  return f;
}

// ---------------- CDNA5 async global->LDS copy (ASYNCcnt path) ----------------

// LDS byte address = low 32 bits of the generic pointer.
__device__ __forceinline__ uint32_t lds_addr32(const void* p) {
  return (uint32_t)(uintptr_t)p;
}

// 16 bytes per lane, memory -> LDS, no VGPR data round-trip.
__device__ __forceinline__ void async_ld16(uint32_t lds_off, const float* g) {
  asm volatile("global_load_async_to_lds_b128 %0, %1, off"
               :: "v"(lds_off), "v"(g)
               : "memory");
}
__device__ __forceinline__ void wait_async0() {
  asm volatile("s_wait_asynccnt 0" ::: "memory");
}

// ---------------- Kernel 0: weight transposes + Ksum zero ----------------

__global__ void k_prep(const float* __restrict__ Wq, const float* __restrict__ Wk,
                       const float* __restrict__ Wv, const float* __restrict__ fcw,
                       float* __restrict__ WTq, float* __restrict__ WTk,
                       float* __restrict__ WTv, float* __restrict__ F1,
                       float* __restrict__ F2, float* __restrict__ Ksum) {
  int i = blockIdx.x * blockDim.x + threadIdx.x;
  if (i < DD * DD) {
    int e = i / DD, d = i % DD;       // W[e][d] -> WT[d][e]
    WTq[d * DD + e] = Wq[i];
    WTk[d * DD + e] = Wk[i];
    WTv[d * DD + e] = Wv[i];
    F1[d * DD + e] = fcw[e * (2 * DD) + d];       // fc_w[e, d]
    F2[d * DD + e] = fcw[e * (2 * DD) + DD + d];  // fc_w[e, D+d]
  }
  if (i < BB * DD) Ksum[i] = 0.0f;
}

// ---------------- Kernel 1: QKV projection + L2 norm + Ksum ----------------

__global__ void __launch_bounds__(256) k_proj(const float* __restrict__ h,
    const float* __restrict__ WTq, const float* __restrict__ WTk,
    const float* __restrict__ WTv, float* __restrict__ Qn, float* __restrict__ Kn,
    float* __restrict__ Vv, float* __restrict__ Ksum) {
  __shared__ float Ht[16 * DD];
  __shared__ float Qt[16 * DD];
  __shared__ float Kt[16 * DD];
  __shared__ float Vt[16 * DD];
  __shared__ float nq[16], nk[16];
  int tid  = threadIdx.x;
  int row0 = blockIdx.x * 16;

  // async-stage the 16x128 h tile into LDS (2 x b128 per thread)
  {
    uint32_t base = lds_addr32(&Ht[0]);
    const float* gsrc = h + (size_t)row0 * DD;
    for (int c = 0; c < 2; ++c) {
      int idx = (tid + 256 * c) * 4;             // float index, 16B chunks
      async_ld16(base + (uint32_t)idx * 4u, gsrc + idx);
    }
  }
  if (tid < 16) { nq[tid] = 0.f; nk[tid] = 0.f; }
  wait_async0();
  __syncthreads();

  int wave = tid >> 5, lane = tid & 31;
  int e0 = wave * 16;
  v8f aq = {}, ak = {}, av = {};
  for (int k = 0; k < DD; k += 4) {
    v2f a = frag_rm(Ht, DD, 0, k, lane);
    aq = wmma_f32(a, frag_cm(WTq, DD, k, e0, lane), aq);
    ak = wmma_f32(a, frag_cm(WTk, DD, k, e0, lane), ak);
    av = wmma_f32(a, frag_cm(WTv, DD, k, e0, lane), av);
  }
  int cc = lane & 15, rbase = (lane >> 4) << 3;
  for (int j = 0; j < 8; ++j) {
    int rr = rbase + j;   // C/D layout: VGPR j -> rows j (lanes 0-15) / j+8 (lanes 16-31)
    Qt[rr * DD + e0 + cc] = aq[j];
    Kt[rr * DD + e0 + cc] = ak[j];
    Vt[rr * DD + e0 + cc] = av[j];
  }
  __syncthreads();

  { // per-row sum of squares: 16 segments of 8 columns per row
    int row = tid & 15, seg = tid >> 4;
    float sq = 0.f, sk = 0.f;
    int base = row * DD + seg * 8;
    for (int c = 0; c < 8; ++c) {
      float q = Qt[base + c]; sq += q * q;
      float kv = Kt[base + c]; sk += kv * kv;
    }
    atomicAdd(&nq[row], sq);
    atomicAdd(&nk[row], sk);
  }
  __syncthreads();
  if (tid < 16) {
    nq[tid] = 1.f / fmaxf(sqrtf(nq[tid]), EPS);
    nk[tid] = 1.f / fmaxf(sqrtf(nk[tid]), EPS);
  }
  __syncthreads();

  for (int i = tid; i < 16 * DD; i += 256) {
    int rr = i >> 7;
    float qn = Qt[i] * nq[rr];
    float kn = Kt[i] * nk[rr];
    Qn[(size_t)row0 * DD + i] = qn;
    Kn[(size_t)row0 * DD + i] = kn;
    Vv[(size_t)row0 * DD + i] = Vt[i];
    Kt[i] = kn;  // keep normalized K for Ksum accumulation
  }
  __syncthreads();
  if (tid < DD) {
    float s = 0.f;
    for (int r = 0; r < 16; ++r) s += Kt[r * DD + tid];
    int b = row0 / (TT * NN);
    atomicAdd(&Ksum[b * DD + tid], s);
  }
}

// ---------------- Kernel 2: M1 = Kn^T V, M2 = Qn^T V (per b,t) ----------------

__global__ void __launch_bounds__(256) k_outer(const float* __restrict__ Qn,
    const float* __restrict__ Kn, const float* __restrict__ Vv,
    float* __restrict__ M1, float* __restrict__ M2) {
  int bt   = blockIdx.x >> 3;
  int sub  = blockIdx.x & 7;
  int wave = threadIdx.x >> 5, lane = threadIdx.x & 31;
  int tile = sub * 8 + wave;             // 0..63 tile of the 128x128 state
  int d1 = (tile >> 3) * 16, d2 = (tile & 7) * 16;
  const float* Kb = Kn + (size_t)bt * NN * DD;
  const float* Qb = Qn + (size_t)bt * NN * DD;
  const float* Vb = Vv + (size_t)bt * NN * DD;
  v8f m1 = {}, m2 = {};
  for (int n = 0; n < NN; n += 4) {
    v2f bv = frag_cm(Vb, DD, n, d2, lane);               // B = V[n, d2..]
    m1 = wmma_f32(frag_cm(Kb, DD, n, d1, lane), bv, m1); // A = Kn^T
    m2 = wmma_f32(frag_cm(Qb, DD, n, d1, lane), bv, m2); // A = Qn^T
  }
  int cc = lane & 15, rbase = (lane >> 4) << 3;
  float* M1b = M1 + (size_t)bt * DD * DD;
  float* M2b = M2 + (size_t)bt * DD * DD;
  for (int j = 0; j < 8; ++j) {
    int rr = d1 + rbase + j;
    M1b[rr * DD + d2 + cc] = m1[j];
    M2b[rr * DD + d2 + cc] = m2[j];
  }
}

// ---- Kernel 3: fused Dn, Qn@M1 / Kn@M2 -> +T*V, *rDn -> concat-FC -> out ----

__global__ void __launch_bounds__(256) k_final(const float* __restrict__ Qn,
    const float* __restrict__ Kn, const float* __restrict__ Vv,
    const float* __restrict__ M1, const float* __restrict__ M2,
    const float* __restrict__ Ksum, const float* __restrict__ F1,
    const float* __restrict__ F2, const float* __restrict__ fcb,
    float* __restrict__ out) {
  __shared__ float Qb[16 * DD], Kb[16 * DD], Ps[16 * DD], Pt[16 * DD];
  __shared__ float ksl[DD];
  __shared__ float rd[16];
  int tid  = threadIdx.x;
  int row0 = blockIdx.x * 16;
  int bt   = row0 / NN;
  int b    = row0 / (TT * NN);

  // async-stage the 16x128 Qn and Kn blocks into LDS
  {
    uint32_t qb_base = lds_addr32(&Qb[0]);
    uint32_t kb_base = lds_addr32(&Kb[0]);
    const float* qs = Qn + (size_t)row0 * DD;
    const float* ks = Kn + (size_t)row0 * DD;
    for (int c = 0; c < 2; ++c) {
      int idx = (tid + 256 * c) * 4;
      async_ld16(qb_base + (uint32_t)idx * 4u, qs + idx);
      async_ld16(kb_base + (uint32_t)idx * 4u, ks + idx);
    }
  }
  if (tid < DD) ksl[tid] = Ksum[b * DD + tid];
  if (tid < 16) rd[tid] = 0.f;
  wait_async0();
  __syncthreads();

  // Dn = Qn . Ksum + T  (segmented dot, 16 segs of 8 per row)
  {
    int row = tid & 15, seg = tid >> 4;
    float p = 0.f;
    int base = row * DD + seg * 8;
    for (int c = 0; c < 8; ++c) p += Qb[base + c] * ksl[seg * 8 + c];
    atomicAdd(&rd[row], p);
  }
  __syncthreads();
  if (tid < 16) rd[tid] = 1.f / ((rd[tid] + (float)TT) * (float)NN);
  __syncthreads();

  int wave = tid >> 5, lane = tid & 31;
  int e0 = wave * 16;
  const float* M1b = M1 + (size_t)bt * DD * DD;
  const float* M2b = M2 + (size_t)bt * DD * DD;
  v8f ps = {}, pt = {};
  for (int k = 0; k < DD; k += 4) {
    ps = wmma_f32(frag_rm(Qb, DD, 0, k, lane), frag_cm(M1b, DD, k, e0, lane), ps);
    pt = wmma_f32(frag_rm(Kb, DD, 0, k, lane), frag_cm(M2b, DD, k, e0, lane), pt);
  }
  int cc = lane & 15, rbase = (lane >> 4) << 3;
  for (int j = 0; j < 8; ++j) {
    int rr = rbase + j;
    float v  = Vv[(size_t)(row0 + rr) * DD + e0 + cc];
    float sc = rd[rr];
    Ps[rr * DD + e0 + cc] = (ps[j] + (float)TT * v) * sc;  // A_s (incl /N via rDn)
    Pt[rr * DD + e0 + cc] = (pt[j] + (float)TT * v) * sc;  // A_t
  }
  __syncthreads();

  v8f acc = {};
  for (int k = 0; k < DD; k += 4) {
    acc = wmma_f32(frag_rm(Ps, DD, 0, k, lane), frag_cm(F1, DD, k, e0, lane), acc);
    acc = wmma_f32(frag_rm(Pt, DD, 0, k, lane), frag_cm(F2, DD, k, e0, lane), acc);
  }
  for (int j = 0; j < 8; ++j) {
    int rr = rbase + j;
    out[(size_t)(row0 + rr) * DD + e0 + cc] = acc[j] + fcb[e0 + cc];
  }
}

// ---------------- host launcher ----------------

extern "C" void kernel_launch(void* const* d_in, const int* in_sizes, int n_in,
                              void* d_out, int out_size, void* d_ws, size_t ws_size,
                              hipStream_t stream) {
  const float* h   = (const float*)d_in[0];
  const float* Wq  = (const float*)d_in[1];
  const float* Wk  = (const float*)d_in[2];
  const float* Wv  = (const float*)d_in[3];
  const float* fcw = (const float*)d_in[4];
  const float* fcb = (const float*)d_in[5];
  float* out = (float*)d_out;

  float* ws  = (float*)d_ws;
  float* WTq = ws;  ws += DD * DD;
  float* WTk = ws;  ws += DD * DD;
  float* WTv = ws;  ws += DD * DD;
  float* F1  = ws;  ws += DD * DD;
  float* F2  = ws;  ws += DD * DD;
  float* Ksum = ws; ws += BB * DD;
  float* Qn = ws;   ws += (size_t)RR * DD;
  float* Kn = ws;   ws += (size_t)RR * DD;
  float* Vv = ws;   ws += (size_t)RR * DD;
  float* M1 = ws;   ws += (size_t)BT * DD * DD;
  float* M2 = ws;   ws += (size_t)BT * DD * DD;

  k_prep<<<(DD * DD + 255) / 256, 256, 0, stream>>>(Wq, Wk, Wv, fcw, WTq, WTk, WTv,
                                                    F1, F2, Ksum);
  k_proj<<<RR / 16, 256, 0, stream>>>(h, WTq, WTk, WTv, Qn, Kn, Vv, Ksum);
  k_outer<<<BT * 8, 256, 0, stream>>>(Qn, Kn, Vv, M1, M2);
  k_final<<<RR / 16, 256, 0, stream>>>(Qn, Kn, Vv, M1, M2, Ksum, F1, F2, fcb, out);
}